// Brown_44513041056401
// MI455X (gfx1250) — compile-verified
//
#include <hip/hip_runtime.h>

// Brown forward: 8-direction masked shift + 3x3 reflect-avg on (32,64,128,128).
// Memory-bound: 512 MiB mandatory traffic -> ~23 us at 23.3 TB/s HBM.
// Strategy: one workgroup per 128x128 plane; stage the plane into LDS via the
// gfx1250 async global->LDS DMA path (ASYNCcnt), fuse dir/prob into 1-byte
// masks in LDS, then each thread replays the sequential d=0..7 write order for
// 4 consecutive pixels and emits a coalesced float4 store.

#define TILE_H 128
#define TILE_W 128
#define PLANE_ELEMS (TILE_H * TILE_W)
#define SMEM_BYTES (PLANE_ELEMS * 4 + PLANE_ELEMS)   // 64KB input + 16KB mask
#define P_THRESH 20

#if defined(__has_builtin)
#if __has_builtin(__builtin_amdgcn_global_load_async_to_lds_b128)
#define BROWN_ASYNC_LDS 1
#endif
#if __has_builtin(__builtin_amdgcn_s_wait_asynccnt)
#define BROWN_WAIT_BUILTIN 1
#endif
#endif

// The builtin's parameters are v4i32 pointers in AS1 (global) / AS3 (LDS),
// per the round-1 diagnostic: 'int __vector(4) __device__ *'.
typedef int brown_v4i __attribute__((ext_vector_type(4)));
typedef __attribute__((address_space(1))) brown_v4i* brown_gv4p;
typedef __attribute__((address_space(3))) brown_v4i* brown_lv4p;

__device__ __forceinline__ int reflect128(int v) {
    // numpy/jax 'reflect' (no edge repeat): -1 -> 1, 128 -> 126. Valid for [-1,128].
    v = (v < 0) ? -v : v;
    v = (v > TILE_H - 1) ? (2 * (TILE_H - 1) - v) : v;
    return v;
}

__global__ void __launch_bounds__(256)
brown_plane_kernel(const float* __restrict__ gin,
                   const int*   __restrict__ gdir,
                   const int*   __restrict__ gprob,
                   float*       __restrict__ gout)
{
    extern __shared__ char smem[];
    float*         s_in   = reinterpret_cast<float*>(smem);
    unsigned char* s_mask = reinterpret_cast<unsigned char*>(smem + PLANE_ELEMS * 4);

    const long long base  = (long long)blockIdx.x * PLANE_ELEMS;
    const float* in_p   = gin  + base;
    const int*   dir_p  = gdir + base;
    const int*   prob_p = gprob + base;
    float*       out_p  = gout + base;
    const int tid = threadIdx.x;

    // ---- Phase 1a: async-DMA the whole input plane into LDS ----------------
#ifdef BROWN_ASYNC_LDS
    #pragma unroll
    for (int k = 0; k < 16; ++k) {
        const int off = (k * 256 + tid) * 4;      // float index, 16B aligned
        __builtin_amdgcn_global_load_async_to_lds_b128(
            (brown_gv4p)(in_p + off), (brown_lv4p)(s_in + off), 0, 0);
    }
#else
    #pragma unroll
    for (int k = 0; k < 16; ++k) {
        const int off = (k * 256 + tid) * 4;
        *reinterpret_cast<float4*>(s_in + off) =
            *reinterpret_cast<const float4*>(in_p + off);
    }
#endif

    // ---- Phase 1b (overlaps DMA): fuse dir/prob -> 1-byte effective dir ----
    // mask = pos ? dir : 8  (8 == "no write"); dir in [0,8)
    #pragma unroll
    for (int k = 0; k < 16; ++k) {
        const int off = (k * 256 + tid) * 4;
        const int4 d4 = *reinterpret_cast<const int4*>(dir_p + off);
        const int4 p4 = *reinterpret_cast<const int4*>(prob_p + off);
        unsigned int m;
        m  = (unsigned int)((p4.x <= P_THRESH) ? d4.x : 8);
        m |= (unsigned int)((p4.y <= P_THRESH) ? d4.y : 8) << 8;
        m |= (unsigned int)((p4.z <= P_THRESH) ? d4.z : 8) << 16;
        m |= (unsigned int)((p4.w <= P_THRESH) ? d4.w : 8) << 24;
        *reinterpret_cast<unsigned int*>(s_mask + off) = m;
    }

#ifdef BROWN_ASYNC_LDS
#ifdef BROWN_WAIT_BUILTIN
    __builtin_amdgcn_s_wait_asynccnt(0);
#else
    asm volatile("s_wait_asynccnt 0x0" ::: "memory");
#endif
#endif
    __syncthreads();

    // ---- Phase 2: replay the sequential d=0..7 write order per pixel -------
    // OFFSETS: 0:(-1,-1) 1:(-1,0) 2:(-1,1) 3:(0,-1) 4:(0,0) 5:(0,1) 6:(1,-1) 7:(1,0)
    constexpr int DI[8] = {-1, -1, -1, 0, 0, 0, 1, 1};
    constexpr int DJ[8] = {-1,  0,  1,-1, 0, 1,-1, 0};

    #pragma unroll 4
    for (int k = 0; k < 16; ++k) {
        const int idx = (k * 256 + tid) * 4;      // 4 consecutive pixels, same row
        const int i   = idx >> 7;
        const int j0  = idx & (TILE_W - 1);
        const int rm  = reflect128(i - 1) * TILE_W;
        const int r0  = i * TILE_W;
        const int rp  = reflect128(i + 1) * TILE_W;

        float res[4];
        #pragma unroll
        for (int t = 0; t < 4; ++t) {
            const int j  = j0 + t;
            const int jl = (j == 0) ? 1 : (j - 1);
            const int jr = (j == TILE_W - 1) ? (TILE_W - 2) : (j + 1);

            // 3x3 reflect-padded box average, reference association order.
            float s = s_in[rm + jl];
            s += s_in[rm + j];  s += s_in[rm + jr];
            s += s_in[r0 + jl]; s += s_in[r0 + j];  s += s_in[r0 + jr];
            s += s_in[rp + jl]; s += s_in[rp + j];  s += s_in[rp + jr];
            const float avg = s / 9.0f;

            float val = s_in[r0 + j];
            const int mself = s_mask[r0 + j];

            #pragma unroll
            for (int d = 0; d < 8; ++d) {
                const int di = DI[d], dj = DJ[d];
                if (d != 4) {
                    // neighbor-write: source (i-di, j-dj) pushes input value here
                    const int si = i - di, sj = j - dj;
                    if ((unsigned)si < (unsigned)TILE_H &&
                        (unsigned)sj < (unsigned)TILE_W &&
                        (int)s_mask[si * TILE_W + sj] == d)
                        val = s_in[si * TILE_W + sj];
                }
                // avg-write at the masked source pixel (target must be in range)
                if (mself == d) {
                    const int ti = i + di, tj = j + dj;
                    if (d == 4 || ((unsigned)ti < (unsigned)TILE_H &&
                                   (unsigned)tj < (unsigned)TILE_W))
                        val = avg;
                }
            }
            res[t] = val;
        }
        float4 v4;
        v4.x = res[0]; v4.y = res[1]; v4.z = res[2]; v4.w = res[3];
        *reinterpret_cast<float4*>(out_p + idx) = v4;
    }
}

extern "C" void kernel_launch(void* const* d_in, const int* in_sizes, int n_in,
                              void* d_out, int out_size, void* d_ws, size_t ws_size,
                              hipStream_t stream) {
    const float* input = (const float*)d_in[0];
    const int*   dir   = (const int*)d_in[1];
    const int*   prob  = (const int*)d_in[2];
    float*       out   = (float*)d_out;

    const int planes = in_sizes[0] / PLANE_ELEMS;   // 32*64 = 2048 planes
    brown_plane_kernel<<<planes, 256, SMEM_BYTES, stream>>>(input, dir, prob, out);

    (void)n_in; (void)out_size; (void)d_ws; (void)ws_size;
}